// AttentionalFactorizationMachine_77601469104538
// MI455X (gfx1250) — compile-verified
//
#include <hip/hip_runtime.h>
#include <math.h>

#define NUM_FIELDS 32
#define EMB_DIM    64
#define ATTN_DIM   64
#define NUM_PAIRS  496
#define NUM_TILES  31      // 496 / 16
#define BATCH      2048

#define XS_STRIDE  66      // padded x row stride (dwords): bank = (2*i + c) % 64
#define W1T_STRIDE 68      // padded W1^T row stride (dwords): bank = (4*n + c) % 64
#define EM_STRIDE  34      // padded attn-matrix row stride (even -> 8B aligned)

typedef float v2f __attribute__((ext_vector_type(2)));
typedef float v8f __attribute__((ext_vector_type(8)));

__global__ __launch_bounds__(256)
void afm_fused_kernel(const float* __restrict__ x,
                      const float* __restrict__ w1,
                      const float* __restrict__ b1,
                      const float* __restrict__ w2,
                      const float* __restrict__ fcw,
                      const float* __restrict__ fcb,
                      float* __restrict__ out)
{
    __shared__ float xs [NUM_FIELDS * XS_STRIDE];   //  8448 B
    __shared__ float w1t[ATTN_DIM   * W1T_STRIDE];  // 17408 B
    __shared__ float b1s[ATTN_DIM];
    __shared__ float w2s[ATTN_DIM];
    __shared__ int   pij[NUM_PAIRS];
    __shared__ float scores[NUM_PAIRS];
    __shared__ float em[NUM_FIELDS * EM_STRIDE];    //  4352 B symmetric attn matrix
    __shared__ float fvec[EMB_DIM];
    __shared__ float red[8];

    const int tid  = threadIdx.x;
    const int lane = tid & 31;
    const int wid  = tid >> 5;          // 0..7
    const int b    = blockIdx.x;

    // ---------------- cooperative staging ----------------
    for (int idx = tid; idx < NUM_FIELDS * EMB_DIM; idx += 256) {
        int f = idx >> 6, d = idx & 63;
        xs[f * XS_STRIDE + d] = x[(size_t)b * (NUM_FIELDS * EMB_DIM) + idx];
    }
    // w1t[a][d] = w1[d][a]  (coalesced global read, transposed LDS write)
    for (int idx = tid; idx < EMB_DIM * ATTN_DIM; idx += 256) {
        int d = idx >> 6, a = idx & 63;
        w1t[a * W1T_STRIDE + d] = w1[idx];
    }
    if (tid < ATTN_DIM) { b1s[tid] = b1[tid]; w2s[tid] = w2[tid]; }
    for (int p = tid; p < NUM_PAIRS; p += 256) {
        int i = 0, rem = p;
        while (rem >= (NUM_FIELDS - 1 - i)) { rem -= (NUM_FIELDS - 1 - i); ++i; }
        pij[p] = (i << 8) | (i + 1 + rem);
    }
    __syncthreads();

    // per-lane constants (C-layout: lane owns column n = lane&15)
    const int n  = lane & 15;
    const int kb = (lane >> 4) << 1;    // 0 or 2 : K-offset of this half-wave
    float b1v[4], w2v[4];
    #pragma unroll
    for (int nb = 0; nb < 4; ++nb) {
        b1v[nb] = b1s[nb * 16 + n];
        w2v[nb] = w2s[nb * 16 + n];
    }

    // ---------------- GEMM 1: scores = relu(xcross @ W1 + b1) @ w2 ----------------
    for (int tile = wid; tile < NUM_TILES; tile += 8) {
        int ij = pij[tile * 16 + n];
        const float* xi = &xs[(ij >> 8)  * XS_STRIDE];
        const float* xj = &xs[(ij & 255) * XS_STRIDE];

        v2f afrag[16];
        #pragma unroll
        for (int ks = 0; ks < 16; ++ks) {
            int c = ks * 4 + kb;                      // even -> 8B aligned
            v2f vi = *(const v2f*)(xi + c);
            v2f vj = *(const v2f*)(xj + c);
            afrag[ks].x = vi.x * vj.x;
            afrag[ks].y = vi.y * vj.y;
        }

        float s[8];
        #pragma unroll
        for (int v = 0; v < 8; ++v) s[v] = 0.0f;

        #pragma unroll
        for (int nb = 0; nb < 4; ++nb) {
            v8f acc = {};
            const float* brow = &w1t[(nb * 16 + n) * W1T_STRIDE + kb];
            #pragma unroll
            for (int ks = 0; ks < 16; ++ks) {
                v2f bfrag = *(const v2f*)(brow + ks * 4);
                acc = __builtin_amdgcn_wmma_f32_16x16x4_f32(
                        false, afrag[ks], false, bfrag,
                        (short)0, acc, false, false);
            }
            #pragma unroll
            for (int v = 0; v < 8; ++v) {             // bias + ReLU + dot w2
                float h = acc[v] + b1v[nb];
                h = h > 0.0f ? h : 0.0f;
                s[v] += h * w2v[nb];
            }
        }
        #pragma unroll
        for (int off = 1; off < 16; off <<= 1) {
            #pragma unroll
            for (int v = 0; v < 8; ++v) s[v] += __shfl_xor(s[v], off, 32);
        }
        if (n == 0) {                                  // lanes 0 and 16 write
            int mbase = tile * 16 + ((lane >> 4) << 3);
            #pragma unroll
            for (int v = 0; v < 8; ++v) scores[mbase + v] = s[v];
        }
    }
    __syncthreads();

    // ---------------- softmax over 496 pairs ----------------
    float lmax = -1e30f;
    for (int p = tid; p < NUM_PAIRS; p += 256) lmax = fmaxf(lmax, scores[p]);
    #pragma unroll
    for (int off = 16; off; off >>= 1) lmax = fmaxf(lmax, __shfl_xor(lmax, off, 32));
    if (lane == 0) red[wid] = lmax;
    if (tid < EMB_DIM) fvec[tid] = 0.0f;
    __syncthreads();

    float gmax = red[0];
    #pragma unroll
    for (int w = 1; w < 8; ++w) gmax = fmaxf(gmax, red[w]);
    __syncthreads();                                   // red about to be reused

    float lsum = 0.0f;
    for (int p = tid; p < NUM_PAIRS; p += 256) {
        float e = __expf(scores[p] - gmax);
        scores[p] = e;                                 // keep unnormalized
        lsum += e;
    }
    #pragma unroll
    for (int off = 16; off; off >>= 1) lsum += __shfl_xor(lsum, off, 32);
    if (lane == 0) red[wid] = lsum;
    __syncthreads();

    float gsum = red[0];
    #pragma unroll
    for (int w = 1; w < 8; ++w) gsum += red[w];
    const float rinv = 1.0f / gsum;

    // ---------------- build symmetric attn matrix A (32x32, zero diag) -------
    if (tid < NUM_FIELDS) em[tid * EM_STRIDE + tid] = 0.0f;
    for (int p = tid; p < NUM_PAIRS; p += 256) {
        int ij = pij[p];
        int i = ij >> 8, j = ij & 255;
        float e = scores[p];
        em[i * EM_STRIDE + j] = e;
        em[j * EM_STRIDE + i] = e;
    }
    __syncthreads();

    // ---------------- GEMM 2: g = A @ x  (32x32 @ 32x64), f = 0.5 * sum_i x_i * g_i
    // one 16x16 C-tile per wave: wave -> (mt = wid&1, nt = wid>>1)
    {
        const int mt = wid & 1;          // M tile (rows i)
        const int nt = wid >> 1;         // N tile (dims d)
        const int m16 = mt * 16 + n;     // this lane's A row
        const int dcol = nt * 16 + n;    // this lane's output column (dim d)

        v8f acc = {};
        #pragma unroll
        for (int ks = 0; ks < 8; ++ks) {
            int c = ks * 4 + kb;
            v2f afrag = *(const v2f*)(&em[m16 * EM_STRIDE + c]);  // A[m][c..c+1]
            v2f bfrag;
            bfrag.x = xs[(c    ) * XS_STRIDE + dcol];             // B[k][d]
            bfrag.y = xs[(c + 1) * XS_STRIDE + dcol];
            acc = __builtin_amdgcn_wmma_f32_16x16x4_f32(
                    false, afrag, false, bfrag, (short)0, acc, false, false);
        }
        // f partial: sum over this lane's 8 rows m = mt*16 + 8*(lane>>4) + v
        const int mbase = mt * 16 + ((lane >> 4) << 3);
        float fp = 0.0f;
        #pragma unroll
        for (int v = 0; v < 8; ++v)
            fp += xs[(mbase + v) * XS_STRIDE + dcol] * acc[v];
        fp += __shfl_xor(fp, 16, 32);                 // combine the two M halves
        if (lane < 16) atomicAdd(&fvec[dcol], fp);    // combine the two mt waves
    }
    __syncthreads();

    // ---------------- y = 0.5 * rinv * (f . fc_w) + fc_b ----------------
    if (wid == 0) {
        float acc = fvec[lane] * fcw[lane] + fvec[lane + 32] * fcw[lane + 32];
        #pragma unroll
        for (int off = 16; off; off >>= 1) acc += __shfl_xor(acc, off, 32);
        if (lane == 0) out[b] = acc * (0.5f * rinv) + fcb[0];
    }
}

extern "C" void kernel_launch(void* const* d_in, const int* in_sizes, int n_in,
                              void* d_out, int out_size, void* d_ws, size_t ws_size,
                              hipStream_t stream) {
    const float* x   = (const float*)d_in[0];
    const float* w1  = (const float*)d_in[1];
    const float* b1  = (const float*)d_in[2];
    const float* w2  = (const float*)d_in[3];
    const float* fcw = (const float*)d_in[4];
    const float* fcb = (const float*)d_in[5];
    float* out = (float*)d_out;

    afm_fused_kernel<<<BATCH, 256, 0, stream>>>(x, w1, b1, w2, fcw, fcb, out);
}